// RGATEncoderLayer_30777735643853
// MI455X (gfx1250) — compile-verified
//
#include <hip/hip_runtime.h>

// ---------------- problem constants ----------------
constexpr int Bc   = 8;
constexpr int Nc   = 256;
constexpr int NFc  = 256;
constexpr int EFc  = 128;
constexpr int Hc   = 8;
constexpr int Oc   = 32;
constexpr int NHID = 1024;
constexpr int HOc  = 256;          // H*O
constexpr float ALPHA = 0.2f;
constexpr float EPSc  = 1e-6f;

typedef __attribute__((ext_vector_type(2))) float v2f;
typedef __attribute__((ext_vector_type(8))) float v8f;

// ---------------- WMMA f32 16x16x4 GEMM core ----------------
// A: row-major MxK tile base (16 rows), lda in elements
// B: row-major KxN tile base (16 cols), ldb in elements
// A frag layout (ISA 7.12.2): lanes 0-15 -> row M=lane, K = {0,1}; lanes 16-31 -> K = {2,3}
// B frag layout (mirror): lanes 0-15 -> col N=lane, K = {0,1}; lanes 16-31 -> K = {2,3}
__device__ __forceinline__ v8f wmma_gemm_f32(const float* __restrict__ A, int lda,
                                             const float* __restrict__ B, int ldb,
                                             int K, v8f acc) {
  const int lane = threadIdx.x & 31;
  const int l16  = lane & 15;
  const int ko   = (lane >> 4) << 1;           // 0 or 2
  const float* ap = A + l16 * lda + ko;
  const float* bp = B + ko * ldb + l16;
#pragma unroll 8
  for (int k = 0; k < K; k += 4) {
    v2f av, bv;
    av.x = ap[k];
    av.y = ap[k + 1];
    bv.x = bp[k * ldb];
    bv.y = bp[k * ldb + ldb];
    acc = __builtin_amdgcn_wmma_f32_16x16x4_f32(false, av, false, bv,
                                                (short)0, acc, false, false);
  }
  return acc;
}

__device__ __forceinline__ float wave_max32(float v) {
#pragma unroll
  for (int m = 16; m >= 1; m >>= 1) v = fmaxf(v, __shfl_xor(v, m, 32));
  return v;
}
__device__ __forceinline__ float wave_sum32(float v) {
#pragma unroll
  for (int m = 16; m >= 1; m >>= 1) v += __shfl_xor(v, m, 32);
  return v;
}

// ---------------- 1) h = einsum('bnd,hdo->bhno') ----------------
__global__ void k_h(const float* __restrict__ x, const float* __restrict__ W,
                    float* __restrict__ h) {
  int wg = blockIdx.x * (blockDim.x >> 5) + (threadIdx.x >> 5);  // 2048 waves
  int nt = wg & 1;             // O tile (0..1)
  int mt = (wg >> 1) & 15;     // N tile (0..15)
  int bh = wg >> 5;            // b*H+head (0..63)
  int b  = bh >> 3, head = bh & 7;
  const float* A  = x + ((long)b * Nc + mt * 16) * NFc;
  const float* Bm = W + (long)head * NFc * Oc + nt * 16;
  v8f acc = {};
  acc = wmma_gemm_f32(A, NFc, Bm, Oc, NFc, acc);
  float* C = h + ((long)bh * Nc + mt * 16) * Oc + nt * 16;
  int lane = threadIdx.x & 31;
  int col  = lane & 15;
  int rb   = (lane >> 4) * 8;
#pragma unroll
  for (int g = 0; g < 8; ++g) C[(rb + g) * Oc + col] = acc[g];
}

// ---------------- 2) s,t = h . a1 / a2 ----------------
__global__ void k_st(const float* __restrict__ h, const float* __restrict__ a,
                     float* __restrict__ s, float* __restrict__ t) {
  int idx  = blockIdx.x * blockDim.x + threadIdx.x;   // b*H*N = 16384
  int head = (idx >> 8) & (Hc - 1);
  const float* hr = h + (long)idx * Oc;
  const float* a1 = a + head * (3 * Oc);
  const float* a2 = a1 + Oc;
  float sv = 0.f, tv = 0.f;
#pragma unroll
  for (int o = 0; o < Oc; ++o) { sv += hr[o] * a1[o]; tv += hr[o] * a2[o]; }
  s[idx] = sv;
  t[idx] = tv;
}

// ---------------- 3) vT[e][hcol] = sum_o W1[h,e,o]*a3[h,o] (padded to 16 cols) ----------------
__global__ void k_v(const float* __restrict__ W1, const float* __restrict__ a,
                    float* __restrict__ vt) {
  int idx = blockIdx.x * blockDim.x + threadIdx.x;    // 2048 = EF*16
  int c = idx & 15, e = idx >> 4;
  float v = 0.f;
  if (c < Hc) {
    const float* w  = W1 + ((long)c * EFc + e) * Oc;
    const float* a3 = a + c * (3 * Oc) + 2 * Oc;
#pragma unroll
    for (int o = 0; o < Oc; ++o) v += w[o] * a3[o];
  }
  vt[idx] = v;
}

// ---------------- 4) u = edge (BN^2 x EF) @ vT (EF x 16) : dominant streaming kernel ----------------
__global__ void k_u(const float* __restrict__ edge, const float* __restrict__ vt,
                    float* __restrict__ u) {
  int wg = blockIdx.x * (blockDim.x >> 5) + (threadIdx.x >> 5);  // 32768 waves
  const float* A = edge + (long)wg * 16 * EFc;
  v8f acc = {};
  acc = wmma_gemm_f32(A, EFc, vt, 16, EFc, acc);
  int lane = threadIdx.x & 31;
  int col  = lane & 15;
  int rb   = (lane >> 4) * 8;
  if (col < Hc) {
#pragma unroll
    for (int g = 0; g < 8; ++g) {
      long row = (long)wg * 16 + rb + g;            // flattened b*N*N + i*N + j
      int b   = (int)(row >> 16);
      int rem = (int)(row & 65535);
      int i = rem >> 8, j = rem & 255;
      u[(((long)b * Hc + col) * Nc + i) * Nc + j] = acc[g];
    }
  }
}

// ---------------- 5) scores -> masked softmax (in place on u) ----------------
__global__ void k_attn(float* __restrict__ u, const float* __restrict__ s,
                       const float* __restrict__ t, const float* __restrict__ adj) {
  int wg   = blockIdx.x * (blockDim.x >> 5) + (threadIdx.x >> 5); // (b,h,i), 16384
  int lane = threadIdx.x & 31;
  int i  = wg & (Nc - 1);
  int bh = wg >> 8;
  int b  = bh >> 3;
  float*       urow = u + (long)wg * Nc;
  const float* trow = t + (long)bh * Nc;
  const float* arow = adj + ((long)b * Nc + i) * Nc;
  float sv = s[wg];
  float vals[8];
  float mx = -3.0e38f;
#pragma unroll
  for (int p = 0; p < 8; ++p) {
    int j = p * 32 + lane;
    float sc = sv + trow[j] + urow[j];
    sc = sc > 0.f ? sc : ALPHA * sc;                 // leaky_relu
    sc = (arow[j] > 0.f) ? sc : -3.0e38f;            // mask
    vals[p] = sc;
    mx = fmaxf(mx, sc);
  }
  mx = wave_max32(mx);
  float sum = 0.f;
#pragma unroll
  for (int p = 0; p < 8; ++p) {
    float e = __expf(vals[p] - mx);
    vals[p] = e;
    sum += e;
  }
  sum = wave_sum32(sum);
  float inv = 1.f / sum;
#pragma unroll
  for (int p = 0; p < 8; ++p) urow[p * 32 + lane] = vals[p] * inv;
}

// ---------------- 6) hp = elu(attn @ h), transposed store to cat[b][i][head*O+o] ----------------
__global__ void k_hp(const float* __restrict__ attn, const float* __restrict__ h,
                     float* __restrict__ cat) {
  int wg = blockIdx.x * (blockDim.x >> 5) + (threadIdx.x >> 5);   // 2048 waves
  int nt = wg & 1;
  int mt = (wg >> 1) & 15;
  int bh = wg >> 5;
  int b  = bh >> 3, head = bh & 7;
  const float* A  = attn + ((long)bh * Nc + mt * 16) * Nc;
  const float* Bm = h + (long)bh * Nc * Oc + nt * 16;
  v8f acc = {};
  acc = wmma_gemm_f32(A, Nc, Bm, Oc, Nc, acc);
  int lane = threadIdx.x & 31;
  int col  = lane & 15;
  int rb   = (lane >> 4) * 8;
#pragma unroll
  for (int g = 0; g < 8; ++g) {
    int i = mt * 16 + rb + g;
    float vv = acc[g];
    vv = vv > 0.f ? vv : (__expf(vv) - 1.f);         // elu
    cat[((long)b * Nc + i) * HOc + head * Oc + nt * 16 + col] = vv;
  }
}

// ---------------- 7/8) generic GEMM + bias (+relu): y = relu(cat@Wo+bo), z = relu(y@w1+b1) ----------------
__global__ void k_mlp(const float* __restrict__ A, const float* __restrict__ Bm,
                      const float* __restrict__ bias, float* __restrict__ C,
                      int Ncols, int K, int relu) {
  int waves_n = Ncols >> 4;
  int wg = blockIdx.x * (blockDim.x >> 5) + (threadIdx.x >> 5);
  int nt = wg % waves_n;
  int mt = wg / waves_n;
  const float* Ap = A + (long)mt * 16 * K;
  const float* Bp = Bm + nt * 16;
  v8f acc = {};
  acc = wmma_gemm_f32(Ap, K, Bp, Ncols, K, acc);
  int lane = threadIdx.x & 31;
  int col  = nt * 16 + (lane & 15);
  int rb   = (lane >> 4) * 8;
#pragma unroll
  for (int g = 0; g < 8; ++g) {
    float vv = acc[g] + bias[col];
    if (relu) vv = fmaxf(vv, 0.f);
    C[((long)(mt * 16 + rb + g)) * Ncols + col] = vv;
  }
}

// ---------------- 9) ff = z @ w2 + b2 + y (residual) ----------------
__global__ void k_ff(const float* __restrict__ z, const float* __restrict__ w2,
                     const float* __restrict__ b2, const float* __restrict__ y,
                     float* __restrict__ ff) {
  int wg = blockIdx.x * (blockDim.x >> 5) + (threadIdx.x >> 5);   // 2048 waves
  int nt = wg & 15;
  int mt = wg >> 4;
  const float* Ap = z + (long)mt * 16 * NHID;
  const float* Bp = w2 + nt * 16;
  v8f acc = {};
  acc = wmma_gemm_f32(Ap, NHID, Bp, HOc, NHID, acc);
  int lane = threadIdx.x & 31;
  int col  = nt * 16 + (lane & 15);
  int rb   = (lane >> 4) * 8;
#pragma unroll
  for (int g = 0; g < 8; ++g) {
    int row = mt * 16 + rb + g;
    ff[(long)row * HOc + col] = acc[g] + b2[col] + y[(long)row * HOc + col];
  }
}

// ---------------- 10) LayerNorm over HO=256, one wave per row ----------------
__global__ void k_ln(const float* __restrict__ ff, const float* __restrict__ g,
                     const float* __restrict__ bb, float* __restrict__ out) {
  int wg   = blockIdx.x * (blockDim.x >> 5) + (threadIdx.x >> 5); // 2048 rows
  int lane = threadIdx.x & 31;
  const float* r = ff + (long)wg * HOc;
  float s1 = 0.f, s2 = 0.f;
#pragma unroll
  for (int p = 0; p < 8; ++p) {
    float v = r[p * 32 + lane];
    s1 += v;
    s2 += v * v;
  }
  s1 = wave_sum32(s1);
  s2 = wave_sum32(s2);
  float mu  = s1 * (1.f / HOc);
  float var = s2 * (1.f / HOc) - mu * mu;
  float inv = rsqrtf(var + EPSc);
  float* o = out + (long)wg * HOc;
#pragma unroll
  for (int p = 0; p < 8; ++p) {
    int c = p * 32 + lane;
    o[c] = (r[c] - mu) * inv * g[c] + bb[c];
  }
}

// ---------------- launch ----------------
extern "C" void kernel_launch(void* const* d_in, const int* in_sizes, int n_in,
                              void* d_out, int out_size, void* d_ws, size_t ws_size,
                              hipStream_t stream) {
  const float* x    = (const float*)d_in[0];
  const float* edge = (const float*)d_in[1];
  const float* adj  = (const float*)d_in[2];
  const float* W    = (const float*)d_in[3];
  const float* W1   = (const float*)d_in[4];
  const float* a    = (const float*)d_in[5];
  const float* Wo   = (const float*)d_in[6];
  const float* bo   = (const float*)d_in[7];
  const float* w1   = (const float*)d_in[8];
  const float* b1   = (const float*)d_in[9];
  const float* w2   = (const float*)d_in[10];
  const float* b2   = (const float*)d_in[11];
  const float* lng  = (const float*)d_in[12];
  const float* lnb  = (const float*)d_in[13];
  float* out = (float*)d_out;

  float* ws  = (float*)d_ws;
  float* h   = ws;                 // 524288
  float* s   = h + 524288;         // 16384
  float* t   = s + 16384;          // 16384
  float* vt  = t + 16384;          // 2048
  float* u   = vt + 2048;          // 4194304
  float* cat = u + 4194304;        // 524288
  float* y   = cat + 524288;       // 524288
  float* z   = y + 524288;         // 2097152
  float* ff  = z + 2097152;        // 524288   (total ~33.7 MB)

  k_h   <<<256,  256, 0, stream>>>(x, W, h);
  k_st  <<<64,   256, 0, stream>>>(h, a, s, t);
  k_v   <<<8,    256, 0, stream>>>(W1, a, vt);
  k_u   <<<4096, 256, 0, stream>>>(edge, vt, u);
  k_attn<<<2048, 256, 0, stream>>>(u, s, t, adj);
  k_hp  <<<256,  256, 0, stream>>>(u, h, cat);
  k_mlp <<<256,  256, 0, stream>>>(cat, Wo, bo, y, HOc, HOc, 1);
  k_mlp <<<1024, 256, 0, stream>>>(y, w1, b1, z, NHID, HOc, 1);
  k_ff  <<<256,  256, 0, stream>>>(z, w2, b2, y, ff);
  k_ln  <<<256,  256, 0, stream>>>(ff, lng, lnb, out);
}